// Operator_89215060672931
// MI455X (gfx1250) — compile-verified
//
#include <hip/hip_runtime.h>

typedef __attribute__((ext_vector_type(2))) float v2f;
typedef __attribute__((ext_vector_type(8))) float v8f;

#define NB 512
#define NT 256

// ---------------------------------------------------------------------------
// Kernel 1: gather + per-element integral + deterministic block reduction.
// out[v] contribution of element e:  det(J_e)/6 * (vals[i0]+vals[i1]+vals[i2])[v]
// ---------------------------------------------------------------------------
__global__ __launch_bounds__(NT) void fem_partial_kernel(
    const float* __restrict__ nodal_values,   // (N_NODES, 8)
    const float* __restrict__ coords,         // (N_NODES, 2)
    const int*   __restrict__ elements,       // (E, 3)
    float* __restrict__ partials,             // (NB, 8) workspace
    int n_elements)
{
    const float2* __restrict__ cptr = (const float2*)coords;
    const float4* __restrict__ vptr = (const float4*)nodal_values; // 2 x float4 per row

    float acc[8];
#pragma unroll
    for (int v = 0; v < 8; ++v) acc[v] = 0.0f;

    const int tid    = threadIdx.x;
    const int stride = gridDim.x * blockDim.x;

    for (int e = blockIdx.x * blockDim.x + tid; e < n_elements; e += stride) {
        const int b  = 3 * e;
        const int i0 = elements[b + 0];
        const int i1 = elements[b + 1];
        const int i2 = elements[b + 2];

        const float2 c0 = cptr[i0];
        const float2 c1 = cptr[i1];
        const float2 c2 = cptr[i2];

        const float ax = c1.x - c0.x, ay = c1.y - c0.y;
        const float bx = c2.x - c0.x, by = c2.y - c0.y;
        const float s  = (ax * by - ay * bx) * (1.0f / 6.0f);

        const float4 p0 = vptr[2 * i0 + 0];
        const float4 p1 = vptr[2 * i0 + 1];
        const float4 q0 = vptr[2 * i1 + 0];
        const float4 q1 = vptr[2 * i1 + 1];
        const float4 r0 = vptr[2 * i2 + 0];
        const float4 r1 = vptr[2 * i2 + 1];

        acc[0] = fmaf(s, p0.x + q0.x + r0.x, acc[0]);
        acc[1] = fmaf(s, p0.y + q0.y + r0.y, acc[1]);
        acc[2] = fmaf(s, p0.z + q0.z + r0.z, acc[2]);
        acc[3] = fmaf(s, p0.w + q0.w + r0.w, acc[3]);
        acc[4] = fmaf(s, p1.x + q1.x + r1.x, acc[4]);
        acc[5] = fmaf(s, p1.y + q1.y + r1.y, acc[5]);
        acc[6] = fmaf(s, p1.z + q1.z + r1.z, acc[6]);
        acc[7] = fmaf(s, p1.w + q1.w + r1.w, acc[7]);
    }

    // Deterministic block tree-reduction, layout [v][tid] -> conflict-free.
    __shared__ float sred[8 * NT];   // 8 KB of the 320 KB/WGP LDS
#pragma unroll
    for (int v = 0; v < 8; ++v) sred[v * NT + tid] = acc[v];
    __syncthreads();

#pragma unroll
    for (int s = NT / 2; s > 0; s >>= 1) {
        if (tid < s) {
#pragma unroll
            for (int v = 0; v < 8; ++v)
                sred[v * NT + tid] += sred[v * NT + tid + s];
        }
        __syncthreads();
    }

    if (tid < 8) partials[blockIdx.x * 8 + tid] = sred[tid * NT];
}

// ---------------------------------------------------------------------------
// Kernel 2: single-wave WMMA reduction of the NB x 8 partials.
// Lane l handles component v = l&7, lane-group g = l>>3 (fixed strided order).
// With A = all-ones, V_WMMA_F32_16X16X4_F32 yields
//   D[m][n] = B_v0[lane n] + B_v1[lane n] + B_v0[lane n+16] + B_v1[lane n+16]
// so with B = {acc, 0}:  D[0][n] = acc[lane n] + acc[lane n+16]
// and out[v] = D[0][v] + D[0][v+8]  (full fp32, fixed order -> deterministic).
// ---------------------------------------------------------------------------
__global__ __launch_bounds__(32) void fem_wmma_reduce_kernel(
    const float* __restrict__ partials,  // (NB, 8)
    float* __restrict__ out,             // (8,)
    int nb)
{
    const int l = threadIdx.x;           // one full wave32, EXEC all ones
    const int v = l & 7;
    const int g = l >> 3;                // 4 lane-groups

    float acc = 0.0f;
    for (int b = g; b < nb; b += 4)      // uniform trip count, no divergence
        acc += partials[b * 8 + v];

    v2f A; A[0] = 1.0f; A[1] = 1.0f;     // ones matrix
    v2f B; B[0] = acc;  B[1] = 0.0f;
    v8f C = {};

    C = __builtin_amdgcn_wmma_f32_16x16x4_f32(
        /*neg_a=*/false, A, /*neg_b=*/false, B,
        /*c_mod=*/(short)0, C, /*reuse_a=*/false, /*reuse_b=*/false);

    __shared__ float smem[32];
    smem[l] = C[0];                      // lane n (n<16) holds D[0][n]
    __syncthreads();

    if (l < 8) out[l] = smem[l] + smem[l + 8];
}

// ---------------------------------------------------------------------------
extern "C" void kernel_launch(void* const* d_in, const int* in_sizes, int n_in,
                              void* d_out, int out_size, void* d_ws, size_t ws_size,
                              hipStream_t stream)
{
    const float* nodal_values = (const float*)d_in[0];
    const float* coords       = (const float*)d_in[1];
    const int*   elements     = (const int*)d_in[2];
    float*       out          = (float*)d_out;
    float*       partials     = (float*)d_ws;       // NB*8 floats = 16 KB

    const int n_elements = in_sizes[2] / 3;

    fem_partial_kernel<<<NB, NT, 0, stream>>>(
        nodal_values, coords, elements, partials, n_elements);

    fem_wmma_reduce_kernel<<<1, 32, 0, stream>>>(partials, out, NB);
}